// QCNN_45930380264053
// MI455X (gfx1250) — compile-verified
//
#include <hip/hip_runtime.h>
#include <math.h>

typedef float v2f __attribute__((ext_vector_type(2)));
typedef float v8f __attribute__((ext_vector_type(8)));

#define NQ  8
#define NST 256

__device__ __forceinline__ v8f wmma_f32_16x16x4(v2f a, v2f b, v8f c) {
  // D = A(16x4) * B(4x16) + C, fp32 matrix pipe (V_WMMA_F32_16X16X4_F32)
  return __builtin_amdgcn_wmma_f32_16x16x4_f32(false, a, false, b, (short)0, c,
                                               false, false);
}

// ---------------------------------------------------------------------------
// Kernel A: build the fixed 256x256 unitary F (conv layers + U3 pooling) by
// applying the gate list to identity columns. 16 blocks x 16 columns each;
// thread t owns amplitude t of all 16 columns in registers, partner exchange
// through bank-padded LDS. Writes Fr, Fi and Fn = -Fi planes (row-major).
// ---------------------------------------------------------------------------
__global__ __launch_bounds__(256)
void build_F_kernel(const float* __restrict__ rzp, const float* __restrict__ ryp,
                    const float* __restrict__ u3p,
                    float* __restrict__ Fr, float* __restrict__ Fi,
                    float* __restrict__ Fn)
{
  __shared__ float sR[NST * 17];
  __shared__ float sI[NST * 17];
  const int t  = threadIdx.x;
  const int c0 = blockIdx.x * 16;

  float pr[16], pi[16];
  #pragma unroll
  for (int j = 0; j < 16; ++j) { pr[j] = (t == c0 + j) ? 1.0f : 0.0f; pi[j] = 0.0f; }

  // generic 2x2 gate: new = diag*own + off*partner   (per-thread coefficients)
  auto gate = [&](int mask, float dr, float di, float orr, float oi) {
    #pragma unroll
    for (int j = 0; j < 16; ++j) { sR[t*17 + j] = pr[j]; sI[t*17 + j] = pi[j]; }
    __syncthreads();
    const int p = t ^ mask;
    #pragma unroll
    for (int j = 0; j < 16; ++j) {
      float br = sR[p*17 + j], bi = sI[p*17 + j];
      float nr = dr*pr[j] - di*pi[j] + orr*br - oi*bi;
      float ni = dr*pi[j] + di*pr[j] + orr*bi + oi*br;
      pr[j] = nr; pi[j] = ni;
    }
    __syncthreads();
  };
  auto cnotg = [&](int wq, int wp) {            // control wq, target wp
    const int cm = 1 << (7 - wq), tm = 1 << (7 - wp);
    #pragma unroll
    for (int j = 0; j < 16; ++j) { sR[t*17 + j] = pr[j]; sI[t*17 + j] = pi[j]; }
    __syncthreads();
    const int src = (t & cm) ? (t ^ tm) : t;
    #pragma unroll
    for (int j = 0; j < 16; ++j) { pr[j] = sR[src*17 + j]; pi[j] = sI[src*17 + j]; }
    __syncthreads();
  };
  auto rzg = [&](float th, int w) {             // diag(e^{-it/2}, e^{+it/2})
    const int m = 1 << (7 - w);
    float c = __cosf(0.5f*th), s = __sinf(0.5f*th);
    gate(m, c, (t & m) ? s : -s, 0.0f, 0.0f);
  };
  auto ryg = [&](float th, int w) {             // [[c,-s],[s,c]]
    const int m = 1 << (7 - w);
    float c = __cosf(0.5f*th), s = __sinf(0.5f*th);
    gate(m, c, 0.0f, (t & m) ? s : -s, 0.0f);
  };
  auto u3g = [&](const float* p3, int w) {
    const int m = 1 << (7 - w);
    float th = p3[0], ph = p3[1], la = p3[2];
    float ctt = __cosf(0.5f*th), stt = __sinf(0.5f*th);
    bool b = (t & m) != 0;
    float dr  = b ? ctt*__cosf(ph + la) :  ctt;
    float di  = b ? ctt*__sinf(ph + la) :  0.0f;
    float orr = b ? stt*__cosf(ph)      : -stt*__cosf(la);
    float oi  = b ? stt*__sinf(ph)      : -stt*__sinf(la);
    gate(m, dr, di, orr, oi);
  };
  auto conv = [&](int i, int wq, int wp) {
    const float* rz = rzp + 3*i;
    const float* ry = ryp + 2*i;
    rzg(rz[0], wq);
    cnotg(wq, wp);
    rzg(rz[1], wp);
    ryg(ry[0], wq);
    cnotg(wp, wq);
    ryg(ry[1], wq);
    cnotg(wq, wp);
    rzg(rz[2], wp);
  };

  conv(0, 1, 0); conv(1, 3, 2); conv(2, 5, 4); conv(3, 7, 6);
  u3g(u3p + 0, 1); u3g(u3p + 3, 3); u3g(u3p + 6, 5); u3g(u3p + 9, 7);
  conv(4, 3, 1); conv(5, 7, 5); conv(6, 5, 3);
  u3g(u3p + 12, 3);
  u3g(u3p + 15, 7);

  #pragma unroll
  for (int j = 0; j < 16; ++j) {
    const int col = c0 + j;
    Fr[t*NST + col] =  pr[j];
    Fi[t*NST + col] =  pi[j];
    Fn[t*NST + col] = -pi[j];
  }
}

// ---------------------------------------------------------------------------
// Kernel C (fused): per block of 16 batch columns:
//   phase 1: 4 cycles of zphase + 7 RXX gates per column, state kept in
//            registers (thread t = amplitude t), partner exchange via LDS;
//            result staged in LDS in B-fragment ([k][n]) layout.
//   phase 2: complex GEMM psi_final = F * Psi via V_WMMA_F32_16X16X4_F32
//            (Re += Fr*Pr + (-Fi)*Pi ; Im += Fr*Pi + Fi*Pr),
//            fused |psi|^2 sign-weighted reduction -> <Z3>, <Z7>,
//            then the 2->10->1 MLP per column.
// ---------------------------------------------------------------------------
__global__ __launch_bounds__(256)
void qcnn_main_kernel(const float* __restrict__ theta, const float* __restrict__ phi,
                      const float* __restrict__ w1, const float* __restrict__ b1,
                      const float* __restrict__ w2, const float* __restrict__ b2,
                      const float* __restrict__ Fr, const float* __restrict__ Fi,
                      const float* __restrict__ Fn,
                      float* __restrict__ out, int Bt)
{
  __shared__ float sR[NST * 9];     // phase-1 exchange scratch (8 cols, pad 9)
  __shared__ float sI[NST * 9];
  __shared__ float sPR[NST * 16];   // Psi_cyc real,  [k][n]
  __shared__ float sPI[NST * 16];   // Psi_cyc imag,  [k][n]
  __shared__ float sZ3[16];
  __shared__ float sZ7[16];

  const int t  = threadIdx.x;
  const int n0 = blockIdx.x * 16;
  if (t < 16) { sZ3[t] = 0.0f; sZ7[t] = 0.0f; }

  const float base = (float)(2 * __popc(t) - NQ);
  // RXX masks: pairs (w,w+1) -> bits (7-w),(6-w); order w = 0,2,4,6,1,3,5
  const int masks[7] = {0xC0, 0x30, 0x0C, 0x03, 0x60, 0x18, 0x06};

  // ---------------- phase 1: two passes of 8 columns ----------------
  for (int pass = 0; pass < 2; ++pass) {
    float pr[8], pi[8], ct[8], st[8], zc[8], zs[8];
    #pragma unroll
    for (int j = 0; j < 8; ++j) {
      int n = n0 + pass*8 + j;
      n = (n < Bt) ? n : (Bt - 1);
      const float th = theta[n], ph = phi[n];
      ct[j] = __cosf(th);       st[j] = __sinf(th);
      const float ang = ph * base;
      zc[j] = __cosf(ang);      zs[j] = __sinf(ang);
      pr[j] = 0.0625f;          pi[j] = 0.0f;     // H^{x8}|0> = 2^{-4}
    }
    __syncthreads();
    for (int cyc = 0; cyc < 4; ++cyc) {
      #pragma unroll
      for (int j = 0; j < 8; ++j) {               // diagonal zphase
        float nr = pr[j]*zc[j] - pi[j]*zs[j];
        float ni = pr[j]*zs[j] + pi[j]*zc[j];
        pr[j] = nr; pi[j] = ni;
      }
      #pragma unroll
      for (int g = 0; g < 7; ++g) {               // psi' = c*psi - i*s*psi[x^m]
        #pragma unroll
        for (int j = 0; j < 8; ++j) { sR[t*9 + j] = pr[j]; sI[t*9 + j] = pi[j]; }
        __syncthreads();
        const int p = t ^ masks[g];
        #pragma unroll
        for (int j = 0; j < 8; ++j) {
          float br = sR[p*9 + j], bi = sI[p*9 + j];
          float nr = ct[j]*pr[j] + st[j]*bi;
          float ni = ct[j]*pi[j] - st[j]*br;
          pr[j] = nr; pi[j] = ni;
        }
        __syncthreads();
      }
    }
    #pragma unroll
    for (int j = 0; j < 8; ++j) {
      sPR[t*16 + pass*8 + j] = pr[j];
      sPI[t*16 + pass*8 + j] = pi[j];
    }
  }
  __syncthreads();

  // ---------------- phase 2: WMMA complex GEMM + reduction ----------------
  const int wv = t >> 5, lane = t & 31;
  const int lm = lane & 15, lh = lane >> 4;

  for (int tt = 0; tt < 2; ++tt) {
    const int m0   = wv*32 + tt*16;
    const int rowA = m0 + lm;
    const float* fr = Fr + rowA*NST;
    const float* fi = Fi + rowA*NST;
    const float* fn = Fn + rowA*NST;
    v8f accR = {0.f,0.f,0.f,0.f,0.f,0.f,0.f,0.f};
    v8f accI = {0.f,0.f,0.f,0.f,0.f,0.f,0.f,0.f};
    for (int k0 = 0; k0 < NST; k0 += 4) {
      const int ka = k0 + 2*lh;                         // A/B: VGPR v <-> K=ka+v
      v2f ar = *(const v2f*)(fr + ka);
      v2f ai = *(const v2f*)(fi + ka);
      v2f an = *(const v2f*)(fn + ka);
      v2f br, bi;
      br.x = sPR[ka*16 + lm];   br.y = sPR[(ka+1)*16 + lm];
      bi.x = sPI[ka*16 + lm];   bi.y = sPI[(ka+1)*16 + lm];
      accR = wmma_f32_16x16x4(ar, br, accR);            // Re += Fr*Pr
      accR = wmma_f32_16x16x4(an, bi, accR);            // Re += (-Fi)*Pi
      accI = wmma_f32_16x16x4(ar, bi, accI);            // Im += Fr*Pi
      accI = wmma_f32_16x16x4(ai, br, accI);            // Im += Fi*Pr
    }
    float a3 = 0.0f, a7 = 0.0f;
    #pragma unroll
    for (int v = 0; v < 8; ++v) {                       // C/D: M = m0+v+8*lh
      const int m = m0 + v + 8*lh;
      const float mag = accR[v]*accR[v] + accI[v]*accI[v];
      a3 += ((m >> 4) & 1) ? -mag : mag;                // wire 3 -> bit 4
      a7 += (m & 1)        ? -mag : mag;                // wire 7 -> bit 0
    }
    atomicAdd(&sZ3[lm], a3);
    atomicAdd(&sZ7[lm], a7);
  }
  __syncthreads();

  // ---------------- tiny MLP head ----------------
  if (t < 16 && (n0 + t) < Bt) {
    const float f0 = sZ3[t], f1 = sZ7[t];
    float acc = b2[0];
    #pragma unroll
    for (int i = 0; i < 10; ++i) {
      float h = tanhf(w1[2*i]*f0 + w1[2*i + 1]*f1 + b1[i]);
      acc += w2[i] * h;
    }
    out[n0 + t] = 1.0f / (1.0f + __expf(-acc));
  }
}

// ---------------------------------------------------------------------------
extern "C" void kernel_launch(void* const* d_in, const int* in_sizes, int n_in,
                              void* d_out, int out_size, void* d_ws, size_t ws_size,
                              hipStream_t stream) {
  (void)n_in; (void)out_size; (void)ws_size;
  const float* theta = (const float*)d_in[0];
  const float* phi   = (const float*)d_in[1];
  const float* rzp   = (const float*)d_in[2];
  const float* ryp   = (const float*)d_in[3];
  const float* u3p   = (const float*)d_in[4];
  const float* w1    = (const float*)d_in[5];
  const float* b1    = (const float*)d_in[6];
  const float* w2    = (const float*)d_in[7];
  const float* b2    = (const float*)d_in[8];
  const int Bt = in_sizes[0];

  float* Fr = (float*)d_ws;          // 256*256 f32 each: Fr, Fi, -Fi
  float* Fi = Fr + NST*NST;
  float* Fn = Fi + NST*NST;

  build_F_kernel<<<16, 256, 0, stream>>>(rzp, ryp, u3p, Fr, Fi, Fn);
  qcnn_main_kernel<<<(Bt + 15) / 16, 256, 0, stream>>>(
      theta, phi, w1, b1, w2, b2, Fr, Fi, Fn, (float*)d_out, Bt);
}